// CATA_89945205113302
// MI455X (gfx1250) — compile-verified
//
#include <hip/hip_runtime.h>
#include <math.h>
#include <stdint.h>

typedef float v2f __attribute__((ext_vector_type(2)));
typedef float v8f __attribute__((ext_vector_type(8)));
typedef unsigned int v4u __attribute__((ext_vector_type(4)));
typedef int v4i __attribute__((ext_vector_type(4)));
typedef int v8i __attribute__((ext_vector_type(8)));

// D = A(16x4 f32) * B(4x16 f32) + C(16x16 f32), full fp32 precision.
#define WMMA_F32(a, b, c) \
  __builtin_amdgcn_wmma_f32_16x16x4_f32(false, (a), false, (b), (short)0, (c), false, false)

#if __has_builtin(__builtin_amdgcn_tensor_load_to_lds) && \
    __has_builtin(__builtin_amdgcn_s_wait_tensorcnt)
#define HAVE_TDM 1
#else
#define HAVE_TDM 0
#endif

namespace {
constexpr int B_ = 2, C_ = 48, H_ = 48, T_ = 400, F_ = 129, W_ = 11;
constexpr int TF = T_ * F_;       // 51600 (divisible by 16)
constexpr int NC = B_ * TF;       // 103200 columns total
constexpr int FP = 144;           // F padded to 9*16
constexpr int NT = NC / 16;       // 6450 column tiles
}

#if HAVE_TDM
// TDM iterate-mode load: nrows rows of 129 fp32, global row stride TF floats,
// LDS destination row stride FP(=144) floats.  Issued once per wave (EXEC-
// independent), tracked on TENSORcnt.
__device__ __forceinline__ void tdm_load_rows(const float* gsrc, float* ldst) {
  const unsigned long long ga = (unsigned long long)(uintptr_t)gsrc;
  const unsigned int laddr = (unsigned int)(uintptr_t)ldst;  // flat addr[31:0] == LDS offset
  // D# group0: count=1, lds_addr, global_addr, type=2 ("image")
  v4u g0 = { 1u,
             laddr,
             (unsigned)(ga & 0xFFFFFFFFu),
             (unsigned)((ga >> 32) & 0x1FFFFFFu) | (2u << 30) };
  // D# group1: data_size=2 (4B), iterate_enable=1; tensor_dim0=129, tensor_dim1=1,
  // tile_dim0=129, tile_dim1=1, tensor_dim0_stride=TF
  v8i g1 = { (int)((2u << 16) | (1u << 19)),
             (int)(129u << 16),          // tensor_dim0[15:0] << 16
             (int)(1u << 16),            // tensor_dim0 hi | tensor_dim1 lo
             (int)(129u << 16),          // tensor_dim1 hi | tile_dim0
             (int)1,                     // tile_dim1=1 | tile_dim2=0
             (int)TF,                    // tensor_dim0_stride lo32
             0, 0 };
  // D# group2 (iterate mode): lds_addr_increment=FP, global_addr_increment=TF,
  // iterate_count = 48-1
  v4i g2 = { 0,
             (int)FP,
             (int)TF,
             (int)(47u << 16) };
  v4i g3 = { 0, 0, 0, 0 };
#if defined(__clang_major__) && __clang_major__ >= 23
  v8i g4 = { 0, 0, 0, 0, 0, 0, 0, 0 };
  __builtin_amdgcn_tensor_load_to_lds(g0, g1, g2, g3, g4, 0);
#else
  __builtin_amdgcn_tensor_load_to_lds(g0, g1, g2, g3, 0);
#endif
}
#endif

// ---------------------------------------------------------------------------
// Pointwise 48x48 conv as GEMM: dst[h,n] = sum_c W[h,c]*(src[c,n]*dw_w[c]+dw_b[c]) + pw_b[h]
// ---------------------------------------------------------------------------
__global__ void pw48_kernel(const float* __restrict__ src, const float* __restrict__ dw_w,
                            const float* __restrict__ dw_b, const float* __restrict__ pw_w,
                            const float* __restrict__ pw_b, float* __restrict__ dst) {
  const int lane = threadIdx.x & 31;
  const int wave = threadIdx.x >> 5;
  const int tile = blockIdx.x * 8 + wave;
  if (tile >= NT) return;
  __builtin_prefetch(pw_w, 0, 1);            // global_prefetch_b8 (weights -> L2)
  const int col0 = tile * 16;
  const int b    = col0 / TF;
  const int tf   = col0 - b * TF + (lane & 15);
  const int half = lane >> 4;
  const int mrow = lane & 15;

  const float* sb = src + (size_t)b * C_ * TF;
  v8f acc0 = {}, acc1 = {}, acc2 = {};
  for (int ks = 0; ks < 12; ++ks) {
    const int k = ks * 4 + half * 2;
    v2f bf;
    bf.x = sb[(size_t)k * TF + tf] * dw_w[k] + dw_b[k];
    bf.y = sb[(size_t)(k + 1) * TF + tf] * dw_w[k + 1] + dw_b[k + 1];
    v2f a0, a1, a2;
    a0.x = pw_w[(mrow)      * 48 + k]; a0.y = pw_w[(mrow)      * 48 + k + 1];
    a1.x = pw_w[(16 + mrow) * 48 + k]; a1.y = pw_w[(16 + mrow) * 48 + k + 1];
    a2.x = pw_w[(32 + mrow) * 48 + k]; a2.y = pw_w[(32 + mrow) * 48 + k + 1];
    acc0 = WMMA_F32(a0, bf, acc0);
    acc1 = WMMA_F32(a1, bf, acc1);
    acc2 = WMMA_F32(a2, bf, acc2);
  }
  float* db = dst + (size_t)b * H_ * TF;
#pragma unroll
  for (int i = 0; i < 8; ++i) {
    const int r = i + half * 8;
    db[(size_t)(r)      * TF + tf] = acc0[i] + pw_b[r];
    db[(size_t)(16 + r) * TF + tf] = acc1[i] + pw_b[16 + r];
    db[(size_t)(32 + r) * TF + tf] = acc2[i] + pw_b[32 + r];
  }
}

// ---------------------------------------------------------------------------
// Per-(b,t) frequency self-attention:
//   S = Q^T K / sqrt(H); A = softmax_g(S); mic_out = X @ A^T
// Q,K,X tiles staged into LDS by the Tensor Data Mover (iterate mode gives the
// 144-float padded row stride directly); S lives in LDS (144x144).
// ---------------------------------------------------------------------------
__global__ void attn_kernel(const float* __restrict__ Q, const float* __restrict__ K,
                            const float* __restrict__ xmic, float* __restrict__ mic_out) {
  extern __shared__ float lds[];
  float* Qs = lds;                 // 48*FP
  float* Ks = Qs + 48 * FP;        // 48*FP
  float* Xs = Ks + 48 * FP;        // 48*FP
  float* Ss = Xs + 48 * FP;        // FP*FP

  const int bt = blockIdx.x;
  const int b  = bt / T_;
  const int t  = bt - b * T_;
  const int tid  = threadIdx.x;
  const int lane = tid & 31;
  const int wave = tid >> 5;
  const int half = lane >> 4;
  const int l15  = lane & 15;

  const float* qb = Q    + (size_t)b * H_ * TF + t * F_;
  const float* kb = K    + (size_t)b * H_ * TF + t * F_;
  const float* xb = xmic + (size_t)b * C_ * TF + t * F_;

#if HAVE_TDM
  // Zero only the 15 pad columns of each 48-row tile while wave 0 drives the TDM.
  for (int i = tid; i < 48 * (FP - F_); i += 256) {
    const int h = i / (FP - F_), f = F_ + i - (i / (FP - F_)) * (FP - F_);
    Qs[h * FP + f] = 0.f; Ks[h * FP + f] = 0.f; Xs[h * FP + f] = 0.f;
  }
  if (wave == 0) {
    tdm_load_rows(qb, Qs);
    tdm_load_rows(kb, Ks);
    tdm_load_rows(xb, Xs);
    __builtin_amdgcn_s_wait_tensorcnt(0);
  }
  __syncthreads();
#else
  for (int i = tid; i < 48 * FP; i += 256) { Qs[i] = 0.f; Ks[i] = 0.f; Xs[i] = 0.f; }
  __syncthreads();
  for (int i = tid; i < 48 * F_; i += 256) {
    const int h = i / F_, f = i - h * F_;
    Qs[h * FP + f] = qb[(size_t)h * TF + f];
    Ks[h * FP + f] = kb[(size_t)h * TF + f];
    Xs[h * FP + f] = xb[(size_t)h * TF + f];
  }
  __syncthreads();
#endif

  // --- S[f,g] = sum_h Q[h,f]*K[h,g] over 9x9 tiles of 16x16, 12 K-steps ---
  for (int tileIdx = wave; tileIdx < 81; tileIdx += 8) {
    const int mt = tileIdx / 9, nt = tileIdx - mt * 9;
    const int fm = mt * 16 + l15;
    const int gn = nt * 16 + l15;
    v8f acc = {};
    for (int ks = 0; ks < 12; ++ks) {
      const int h = ks * 4 + half * 2;
      v2f a, bf;
      a.x  = Qs[h * FP + fm]; a.y  = Qs[(h + 1) * FP + fm];
      bf.x = Ks[h * FP + gn]; bf.y = Ks[(h + 1) * FP + gn];
      acc = WMMA_F32(a, bf, acc);
    }
#pragma unroll
    for (int i = 0; i < 8; ++i)
      Ss[(mt * 16 + i + half * 8) * FP + nt * 16 + l15] = acc[i];
  }
  __syncthreads();

  // --- softmax over g (g<129), one thread per row f ---
  if (tid < F_) {
    float* row = Ss + tid * FP;
    const float scale = 0.14433756729740643f;  // 1/sqrt(48)
    float m = -__builtin_inff();
    for (int g = 0; g < F_; ++g) m = fmaxf(m, row[g] * scale);
    float s = 0.f;
    for (int g = 0; g < F_; ++g) { float e = __expf(row[g] * scale - m); row[g] = e; s += e; }
    const float inv = 1.f / s;
    for (int g = 0; g < F_; ++g) row[g] *= inv;
    // cols g in [129,144) stayed 0 from the GEMM (zero K pad) -> safe as K-pad below
  }
  __syncthreads();

  // --- mic_out[c,f] = sum_g X[c,g]*A[f,g] : M=48, N=144(tiles), K=132 ---
  float* mo = mic_out + (size_t)b * C_ * TF + t * F_;
  for (int tileIdx = wave; tileIdx < 27; tileIdx += 8) {
    const int mt = tileIdx / 9, nt = tileIdx - mt * 9;
    const int cm = mt * 16 + l15;
    const int fn = nt * 16 + l15;
    v8f acc = {};
    for (int ks = 0; ks < 33; ++ks) {
      const int g = ks * 4 + half * 2;
      v2f a, bf;
      a.x  = Xs[cm * FP + g]; a.y  = Xs[cm * FP + g + 1];
      bf.x = Ss[fn * FP + g]; bf.y = Ss[fn * FP + g + 1];
      acc = WMMA_F32(a, bf, acc);
    }
#pragma unroll
    for (int i = 0; i < 8; ++i) {
      const int c = mt * 16 + i + half * 8;
      if (fn < F_) mo[(size_t)c * TF + fn] = acc[i];
    }
  }
}

// ---------------------------------------------------------------------------
// Windowed causal cross-attention with reference signal (pure VALU, wave32).
// ---------------------------------------------------------------------------
__global__ void ref_attn_kernel(const float* __restrict__ Q, const float* __restrict__ Kref,
                                const float* __restrict__ xref, float* __restrict__ ref_out) {
  const int bt = blockIdx.x;
  const int b  = bt / T_;
  const int t  = bt - b * T_;
  const int lane = threadIdx.x & 31;
  const int wave = threadIdx.x >> 5;

  for (int c = wave; c < C_; c += 8) {
    const float* qrow = Q + ((size_t)(b * H_ + c) * T_ + t) * F_;
    float q[5], acc[5];
#pragma unroll
    for (int i = 0; i < 5; ++i) {
      const int f = lane + 32 * i;
      q[i] = (f < F_) ? qrow[f] : 0.f;
      acc[i] = 0.f;
    }
    for (int j = 0; j < W_; ++j) {
      const int tj = t + j - (W_ - 1);
      if (tj < 0) continue;   // zero-padded window: contributes exactly 0
      const float* krow = Kref + ((size_t)(b * H_ + c) * T_ + tj) * F_;
      const float* vrow = xref + ((size_t)(b * C_ + c) * T_ + tj) * F_;
      float p[5], v[5];
      float m = -__builtin_inff();
#pragma unroll
      for (int i = 0; i < 5; ++i) {
        const int f = lane + 32 * i;
        if (f < F_) { p[i] = q[i] * krow[f]; v[i] = vrow[f]; }
        else        { p[i] = -__builtin_inff(); v[i] = 0.f; }
        m = fmaxf(m, p[i]);
      }
      for (int off = 16; off > 0; off >>= 1) m = fmaxf(m, __shfl_xor(m, off, 32));
      float e[5], s = 0.f;
#pragma unroll
      for (int i = 0; i < 5; ++i) {
        const int f = lane + 32 * i;
        e[i] = (f < F_) ? __expf(p[i] - m) : 0.f;
        s += e[i];
      }
      for (int off = 16; off > 0; off >>= 1) s += __shfl_xor(s, off, 32);
      const float inv = 1.f / s;
#pragma unroll
      for (int i = 0; i < 5; ++i) acc[i] += v[i] * e[i] * inv;
    }
    float* orow = ref_out + ((size_t)(b * C_ + c) * T_ + t) * F_;
#pragma unroll
    for (int i = 0; i < 5; ++i) {
      const int f = lane + 32 * i;
      if (f < F_) orow[f] = acc[i];
    }
  }
}

// ---------------------------------------------------------------------------
// Fusion: 96x96 GEMM over concat([mic_out, ref_out]) + BatchNorm(eval) + PReLU
// ---------------------------------------------------------------------------
__global__ void fusion_kernel(const float* __restrict__ mic, const float* __restrict__ ref,
                              const float* __restrict__ fw, const float* __restrict__ fb,
                              const float* __restrict__ gamma, const float* __restrict__ beta,
                              const float* __restrict__ mean, const float* __restrict__ var,
                              const float* __restrict__ prelu_a, float* __restrict__ out) {
  const int lane = threadIdx.x & 31;
  const int wave = threadIdx.x >> 5;
  const int tile = blockIdx.x * 8 + wave;
  if (tile >= NT) return;
  __builtin_prefetch(fw, 0, 1);
  const int col0 = tile * 16;
  const int b    = col0 / TF;
  const int tf   = col0 - b * TF + (lane & 15);
  const int half = lane >> 4;
  const int mrow = lane & 15;
  const float alpha = prelu_a[0];

  const float* micb = mic + (size_t)b * C_ * TF;
  const float* refb = ref + (size_t)b * C_ * TF;
  v8f acc[6] = {};
  for (int ks = 0; ks < 24; ++ks) {
    const int k = ks * 4 + half * 2;
    v2f bf;
    bf.x = (k < 48)     ? micb[(size_t)k * TF + tf]       : refb[(size_t)(k - 48) * TF + tf];
    bf.y = (k + 1 < 48) ? micb[(size_t)(k + 1) * TF + tf] : refb[(size_t)(k - 47) * TF + tf];
#pragma unroll
    for (int mt = 0; mt < 6; ++mt) {
      v2f a;
      const int r = mt * 16 + mrow;
      a.x = fw[r * 96 + k]; a.y = fw[r * 96 + k + 1];
      acc[mt] = WMMA_F32(a, bf, acc[mt]);
    }
  }
  float* ob = out + (size_t)b * 96 * TF;
#pragma unroll
  for (int mt = 0; mt < 6; ++mt) {
#pragma unroll
    for (int i = 0; i < 8; ++i) {
      const int o = mt * 16 + i + half * 8;
      const float s = gamma[o] * rsqrtf(var[o] + 1e-5f);
      const float y = (acc[mt][i] + fb[o] - mean[o]) * s + beta[o];
      ob[(size_t)o * TF + tf] = (y >= 0.f) ? y : alpha * y;
    }
  }
}

// ---------------------------------------------------------------------------
extern "C" void kernel_launch(void* const* d_in, const int* in_sizes, int n_in,
                              void* d_out, int out_size, void* d_ws, size_t ws_size,
                              hipStream_t stream) {
  (void)in_sizes; (void)n_in; (void)out_size; (void)ws_size;
  const float* x_mic  = (const float*)d_in[0];
  const float* x_ref  = (const float*)d_in[1];
  const float* dw1_w  = (const float*)d_in[2];
  const float* dw1_b  = (const float*)d_in[3];
  const float* pw1_w  = (const float*)d_in[4];
  const float* pw1_b  = (const float*)d_in[5];
  const float* dw2_w  = (const float*)d_in[6];
  const float* dw2_b  = (const float*)d_in[7];
  const float* pw2_w  = (const float*)d_in[8];
  const float* pw2_b  = (const float*)d_in[9];
  const float* dw3_w  = (const float*)d_in[10];
  const float* dw3_b  = (const float*)d_in[11];
  const float* pw3_w  = (const float*)d_in[12];
  const float* pw3_b  = (const float*)d_in[13];
  const float* fus_w  = (const float*)d_in[14];
  const float* fus_b  = (const float*)d_in[15];
  const float* bn_g   = (const float*)d_in[16];
  const float* bn_b   = (const float*)d_in[17];
  const float* bn_m   = (const float*)d_in[18];
  const float* bn_v   = (const float*)d_in[19];
  const float* prelua = (const float*)d_in[20];
  // d_in[21] delay_factor (unused: 0 -> window=W), d_in[22] window (constant 11)

  float* out     = (float*)d_out;                           // (B,96,T,F)
  float* mic_out = out + (size_t)B_ * 96 * TF;              // (B,48,T,F) second output
  float* Q    = (float*)d_ws;
  float* K    = Q    + (size_t)B_ * H_ * TF;
  float* Kref = K    + (size_t)B_ * H_ * TF;
  float* refO = Kref + (size_t)B_ * H_ * TF;

  const dim3 blk(256);
  const int gemmBlocks = (NT + 7) / 8;                      // 807

  pw48_kernel<<<gemmBlocks, blk, 0, stream>>>(x_mic, dw1_w, dw1_b, pw1_w, pw1_b, Q);
  pw48_kernel<<<gemmBlocks, blk, 0, stream>>>(Q,     dw2_w, dw2_b, pw2_w, pw2_b, K);
  pw48_kernel<<<gemmBlocks, blk, 0, stream>>>(x_ref, dw3_w, dw3_b, pw3_w, pw3_b, Kref);

  const size_t ldsBytes = (size_t)(3 * 48 * FP + FP * FP) * sizeof(float);  // ~162 KB < 320 KB/WGP
  attn_kernel<<<B_ * T_, blk, ldsBytes, stream>>>(Q, K, x_mic, mic_out);

  ref_attn_kernel<<<B_ * T_, blk, 0, stream>>>(Q, Kref, x_ref, refO);

  fusion_kernel<<<gemmBlocks, blk, 0, stream>>>(mic_out, refO, fus_w, fus_b,
                                                bn_g, bn_b, bn_m, bn_v, prelua, out);
}